// Relation_30580167147772
// MI455X (gfx1250) — compile-verified
//
#include <hip/hip_runtime.h>
#include <math.h>

// Problem constants (match reference)
constexpr int Np  = 2048;  // proposals
constexpr int Dp  = 128;   // appearance dim
constexpr int DKp = 64;    // key/query dim
constexpr int DGp = 64;    // geometric dim
constexpr int SDp = 8;     // relation output dim

typedef __attribute__((ext_vector_type(2))) float v2f;
typedef __attribute__((ext_vector_type(8))) float v8f;

// ---------------------------------------------------------------- kernel 0
__global__ void zero_colsum_kernel(float* __restrict__ colsum) {
    int i = blockIdx.x * blockDim.x + threadIdx.x;
    if (i < Np) colsum[i] = 0.0f;
}

// ---------------------------------------------------------------- kernel 1a
// K and Q projections fused: thread (i, c) with c<64 -> K col c, c>=64 -> Q col c-64
__global__ __launch_bounds__(256) void proj_kq_kernel(
    const float* __restrict__ A,
    const float* __restrict__ Wk, const float* __restrict__ bk,
    const float* __restrict__ Wq, const float* __restrict__ bq,
    float* __restrict__ Kp, float* __restrict__ Qp) {
    int idx = blockIdx.x * 256 + threadIdx.x;          // Np*128 threads
    int i = idx >> 7;
    int c = idx & 127;
    bool isQ = (c >= 64);
    int cc = c & 63;
    const float* __restrict__ Arow = A + (size_t)i * Dp;
    const float* __restrict__ W = (isQ ? Wq : Wk) + cc; // column cc, row stride DKp
    float acc = isQ ? bq[cc] : bk[cc];
#pragma unroll 8
    for (int d = 0; d < Dp; ++d) acc = fmaf(Arow[d], W[(size_t)d * DKp], acc);
    (isQ ? Qp : Kp)[(size_t)i * DKp + cc] = acc;
}

// ---------------------------------------------------------------- kernel 1b
__global__ __launch_bounds__(256) void proj_v_kernel(
    const float* __restrict__ A, const float* __restrict__ Wv,
    const float* __restrict__ bv, float* __restrict__ Vp) {
    int idx = blockIdx.x * 256 + threadIdx.x;          // Np*SDp threads
    int i = idx >> 3;
    int s = idx & 7;
    const float* __restrict__ Arow = A + (size_t)i * Dp;
    float acc = bv[s];
#pragma unroll 8
    for (int d = 0; d < Dp; ++d) acc = fmaf(Arow[d], Wv[(size_t)d * SDp + s], acc);
    Vp[(size_t)i * SDp + s] = acc;
}

// ---------------------------------------------------------------- kernel 2
// Fused: per 16x16 tile (one wave32 each):
//   w_A tile = K * Q^T via chained V_WMMA_F32_16X16X4_F32 (16 k-steps, DK=64)
//   w_G      = relu(geom . Wg + bg) streamed from HBM (the dominant 1.07 GB read)
//   w_nom    = w_G * exp(w_A / 8)  -> stored to ws (L2-resident 16 MB)
//   colsum  += per-column partial sums (wave shuffle reduce + f32 atomics)
// Wg/bg are wave-uniform -> read through scalar loads (SGPR-resident), keeping
// VGPRs free for the geom streaming window.
__global__ __launch_bounds__(256) void relation_main_kernel(
    const float* __restrict__ Kp, const float* __restrict__ Qp,
    const float* __restrict__ geom,
    const float* __restrict__ WgG, const float* __restrict__ bgG,
    float* __restrict__ wnom, float* __restrict__ colsum) {

    const int lane = threadIdx.x & 31;
    const int wave = threadIdx.x >> 5;
    const int tile = blockIdx.x * 8 + wave;    // 128x128 tiles total
    const int ti = (tile >> 7) << 4;           // tile row origin (i)
    const int tj = (tile & 127) << 4;          // tile col origin (j)
    const int m    = lane & 15;                // M (for A) / N (for B,C,D)
    const int half = lane >> 4;                // lane group 0/1
    const int jj = tj + m;

    // Kick the HBM fetch of each lane's first geom row before the WMMA phase
    // (256 B per (i,j) pair = 2 cachelines).
    {
        const float* g0 = geom + ((size_t)(ti + half * 8) * Np + jj) * DGp;
        __builtin_prefetch(g0, 0, 1);
        __builtin_prefetch(g0 + 32, 0, 1);
    }

    // ISA f32 16x4 A layout: VGPR0 = K{0|2}, VGPR1 = K{1|3} (halves by lane group)
    const float* __restrict__ Ka = Kp + (size_t)(ti + m) * DKp + half * 2;
    const float* __restrict__ Qb = Qp + (size_t)(tj + m) * DKp + half * 2;

    v8f c = {};
#pragma unroll
    for (int t = 0; t < 16; ++t) {
        v2f a = *(const v2f*)(Ka + 4 * t);
        v2f b = *(const v2f*)(Qb + 4 * t);
        // 8 args: (neg_a, A, neg_b, B, c_mod, C, reuse_a, reuse_b)
        c = __builtin_amdgcn_wmma_f32_16x16x4_f32(
                false, a, false, b, (short)0, c, false, false);
    }

    // C/D layout: element c[r] of lane -> (i = ti + r + 8*half, j = tj + m)
    const float kScale = 0.125f;   // 1/sqrt(DK)
    const float bgv = bgG[0];      // uniform -> SGPR
    float csum = 0.0f;
#pragma unroll 4
    for (int r = 0; r < 8; ++r) {
        const int ii = ti + r + half * 8;
        const float* __restrict__ g = geom + ((size_t)ii * Np + jj) * DGp;
        // two accumulators for dual-issue FMA chains
        float wg0 = bgv, wg1 = 0.0f;
#pragma unroll
        for (int d = 0; d < DGp; d += 8) {
            float4 ga = *(const float4*)(g + d);
            float4 gb = *(const float4*)(g + d + 4);
            // WgG[] indices are compile-time -> uniform scalar loads (SGPRs)
            wg0 = fmaf(ga.x, WgG[d + 0], wg0);
            wg1 = fmaf(ga.y, WgG[d + 1], wg1);
            wg0 = fmaf(ga.z, WgG[d + 2], wg0);
            wg1 = fmaf(ga.w, WgG[d + 3], wg1);
            wg0 = fmaf(gb.x, WgG[d + 4], wg0);
            wg1 = fmaf(gb.y, WgG[d + 5], wg1);
            wg0 = fmaf(gb.z, WgG[d + 6], wg0);
            wg1 = fmaf(gb.w, WgG[d + 7], wg1);
        }
        float wg = fmaxf(wg0 + wg1, 0.0f);
        float val = wg * __expf(c[r] * kScale);
        wnom[(size_t)ii * Np + jj] = val;
        csum += val;
    }
    // column partials: rows M=0..7 in lanes 0-15, M=8..15 in lanes 16-31
    csum += __shfl_xor(csum, 16, 32);
    if (half == 0) atomicAdd(&colsum[jj], csum);
}

// ---------------------------------------------------------------- kernel 3
__global__ void scale_v_kernel(const float* __restrict__ Vp,
                               const float* __restrict__ colsum,
                               float* __restrict__ Vs) {
    int idx = blockIdx.x * blockDim.x + threadIdx.x;   // Np*SDp
    if (idx < Np * SDp) {
        int j = idx >> 3;
        Vs[idx] = Vp[idx] / colsum[j];
    }
}

// ---------------------------------------------------------------- kernel 4
// out[i, 0..7] = sum_j wnom[i][j] * Vs[j][0..7]   (wnom is L2-resident)
__global__ __launch_bounds__(64) void out_gemv_kernel(
    const float* __restrict__ wnom, const float* __restrict__ Vs,
    float* __restrict__ out) {
    __shared__ float red[64][SDp + 1];   // +1 pad: no LDS bank conflicts
    const int i = blockIdx.x;
    const int t = threadIdx.x;
    const float* __restrict__ wrow = wnom + (size_t)i * Np;
    float acc[SDp] = {};
    for (int j = t; j < Np; j += 64) {
        float w = wrow[j];
        const float4 v0 = *(const float4*)(Vs + (size_t)j * SDp);
        const float4 v1 = *(const float4*)(Vs + (size_t)j * SDp + 4);
        acc[0] = fmaf(w, v0.x, acc[0]); acc[1] = fmaf(w, v0.y, acc[1]);
        acc[2] = fmaf(w, v0.z, acc[2]); acc[3] = fmaf(w, v0.w, acc[3]);
        acc[4] = fmaf(w, v1.x, acc[4]); acc[5] = fmaf(w, v1.y, acc[5]);
        acc[6] = fmaf(w, v1.z, acc[6]); acc[7] = fmaf(w, v1.w, acc[7]);
    }
#pragma unroll
    for (int s = 0; s < SDp; ++s) red[t][s] = acc[s];
    __syncthreads();
    if (t < SDp) {
        float s_ = 0.0f;
#pragma unroll 8
        for (int u = 0; u < 64; ++u) s_ += red[u][t];
        out[(size_t)i * SDp + t] = s_;
    }
}

// ---------------------------------------------------------------- launch
extern "C" void kernel_launch(void* const* d_in, const int* in_sizes, int n_in,
                              void* d_out, int out_size, void* d_ws, size_t ws_size,
                              hipStream_t stream) {
    (void)in_sizes; (void)n_in; (void)out_size; (void)ws_size;
    const float* A    = (const float*)d_in[0];
    const float* geom = (const float*)d_in[1];
    const float* Wv   = (const float*)d_in[2];
    const float* bv   = (const float*)d_in[3];
    const float* Wk   = (const float*)d_in[4];
    const float* bk   = (const float*)d_in[5];
    const float* Wq   = (const float*)d_in[6];
    const float* bq   = (const float*)d_in[7];
    const float* Wg   = (const float*)d_in[8];
    const float* bg   = (const float*)d_in[9];
    float* out = (float*)d_out;

    // workspace layout (floats): K | Q | V | Vscaled | colsum | wnom  (~18 MB)
    float* ws     = (float*)d_ws;
    float* Kp     = ws;                       // Np*DKp
    float* Qp     = Kp + (size_t)Np * DKp;    // Np*DKp
    float* Vp     = Qp + (size_t)Np * DKp;    // Np*SDp
    float* Vs     = Vp + (size_t)Np * SDp;    // Np*SDp
    float* colsum = Vs + (size_t)Np * SDp;    // Np
    float* wnom   = colsum + Np;              // Np*Np (16 MB, L2-resident)

    zero_colsum_kernel<<<(Np + 255) / 256, 256, 0, stream>>>(colsum);
    proj_kq_kernel<<<(Np * 128) / 256, 256, 0, stream>>>(A, Wk, bk, Wq, bq, Kp, Qp);
    proj_v_kernel<<<(Np * SDp) / 256, 256, 0, stream>>>(A, Wv, bv, Vp);
    relation_main_kernel<<<(128 * 128) / 8, 256, 0, stream>>>(
        Kp, Qp, geom, Wg, bg, wnom, colsum);
    scale_v_kernel<<<(Np * SDp + 255) / 256, 256, 0, stream>>>(Vp, colsum, Vs);
    out_gemv_kernel<<<Np, 64, 0, stream>>>(wnom, Vs, out);
}